// StatePolicy_43344809952098
// MI455X (gfx1250) — compile-verified
//
#include <hip/hip_runtime.h>
#include <math.h>

typedef __attribute__((ext_vector_type(2))) float v2f;
typedef __attribute__((ext_vector_type(8))) float v8f;

#define NL   1000000
#define TILE 250          // output links per block
#define EXT  256          // TILE + 2*3 halo
#define HSTR 258          // EXT + 2 zero guard cells
#define NCH  7

__global__ __launch_bounds__(256) void
state_policy_kernel(const float* __restrict__ x,
                    const float* __restrict__ states,
                    const float* __restrict__ conv_w,
                    const float* __restrict__ conv_b,
                    const float* __restrict__ comp_mat,
                    const float* __restrict__ act_mat,
                    float* __restrict__ out)
{
    __shared__ float h_lds[NCH * HSTR];   // h with 1-cell zero guards on both ends
    __shared__ float st_lds[NCH * EXT];   // recurrent state for the extended tile

    const int tid  = threadIdx.x;         // 0..255, owns extended link l0-3+tid
    const int lane = tid & 31;
    const int wid  = tid >> 5;            // 0..7 waves
    const int n    = lane & 15;           // WMMA row/col within 16
    const int hi   = lane >> 4;           // 0: K pair {0,1}; 1: K pair {2,3}

    const long l0    = (long)blockIdx.x * TILE;
    const long e     = l0 - 3 + tid;      // extended global link index
    const bool valid = (e >= 0) && (e < NL);

    // ---- per-thread static inputs (coalesced along links) ----
    float jl[6], jd[6];
#pragma unroll
    for (int c = 0; c < 6; ++c) {
        long flat = (long)c * NL + e;
        bool ok = valid && (flat >= 2);
        jl[c] = ok ? x[7 + flat - 2]            : 0.0f;
        jd[c] = ok ? x[6L * NL + 11 + flat - 2] : 0.0f;
    }
#pragma unroll
    for (int c = 0; c < NCH; ++c)
        st_lds[c * EXT + tid] = valid ? states[(long)c * NL + e] : 0.0f;

    const float ext0 = x[3];
    const float ext1 = x[4];

    float comp[NCH][3];
#pragma unroll
    for (int c = 0; c < NCH; ++c)
#pragma unroll
        for (int j = 0; j < 3; ++j)
            comp[c][j] = comp_mat[c * 3 + j];

    float cb[NCH];
#pragma unroll
    for (int o = 0; o < NCH; ++o) cb[o] = conv_b[o];

    // ---- A operand: W (7x21 im2col of conv_w, zero-padded to 16x24) ----
    // 32-bit A 16x4 layout: lanes 0-15 hold M=lane, K={0,1}; lanes 16-31 hold M=lane-16, K={2,3}
    v2f A[6];
#pragma unroll
    for (int kb = 0; kb < 6; ++kb) {
        int k0 = kb * 4 + 2 * hi;
        float ax = 0.0f, ay = 0.0f;
        if (n < NCH) {
            if (k0 < 21)     ax = conv_w[n * 21 + k0];
            if (k0 + 1 < 21) ay = conv_w[n * 21 + k0 + 1];
        }
        A[kb].x = ax; A[kb].y = ay;
    }

    // permanent zero guards for conv padding
    if (tid < NCH) {
        h_lds[tid * HSTR + 0]        = 0.0f;
        h_lds[tid * HSTR + HSTR - 1] = 0.0f;
    }
    __syncthreads();

    for (int it = 0; it < 3; ++it) {
        // ---- phase A: elementwise gate h = tanh(obs0*c0 + obs1*c1 + st*c2) ----
#pragma unroll
        for (int c = 0; c < NCH; ++c) {
            float o0 = (c < 6) ? jl[c] : ext0;
            float o1 = (c < 6) ? jd[c] : ext1;
            float sv = st_lds[c * EXT + tid];
            float hv = tanhf(o0 * comp[c][0] + o1 * comp[c][1] + sv * comp[c][2]);
            h_lds[c * HSTR + tid + 1] = valid ? hv : 0.0f;
        }
        __syncthreads();

        // ---- phase B: 3-tap conv as 16x16x24 GEMM via chained f32 WMMA ----
        // c_out[o,l] = sum_kappa W[o,kappa] * h[kappa/3, l + kappa%3 - 1]
#pragma unroll
        for (int gi = 0; gi < 2; ++gi) {
            int base = (wid * 2 + gi) * 16;   // uniform per wave
            v8f acc = {0.f, 0.f, 0.f, 0.f, 0.f, 0.f, 0.f, 0.f};
#pragma unroll
            for (int kb = 0; kb < 6; ++kb) {
                int k0 = kb * 4 + 2 * hi;
                v2f B;
                {
                    bool vx = (k0 < 21);
                    int ii  = vx ? (k0 / 3) : 0;
                    int kk  = vx ? (k0 % 3) : 0;
                    float r = h_lds[ii * HSTR + base + n + kk];
                    B.x = vx ? r : 0.0f;
                }
                {
                    int k1 = k0 + 1;
                    bool vy = (k1 < 21);
                    int ii  = vy ? (k1 / 3) : 0;
                    int kk  = vy ? (k1 % 3) : 0;
                    float r = h_lds[ii * HSTR + base + n + kk];
                    B.y = vy ? r : 0.0f;
                }
                acc = __builtin_amdgcn_wmma_f32_16x16x4_f32(
                        false, A[kb], false, B, (short)0, acc, false, false);
            }
            // D layout: lane n<16 holds out[o=v, link base+n] in acc[v], v=0..7 (o<7 valid)
            if (lane < 16) {
#pragma unroll
                for (int o = 0; o < NCH; ++o)
                    st_lds[o * EXT + base + n] = tanhf(acc[o] + cb[o]);
            }
        }
        __syncthreads();
    }

    // ---- phase C: acts = am0*st + am1*jdlrs, shifted store ----
    if (tid >= 3 && tid < 3 + TILE) {
        long l = l0 + tid - 3;
        if (l < NL) {
#pragma unroll
            for (int c = 0; c < 6; ++c) {
                float a0  = act_mat[c * 2 + 0];
                float a1  = act_mat[c * 2 + 1];
                float stv = st_lds[c * EXT + tid];
                long idx  = (long)c * NL + l - 2;
                if (idx >= 0) out[idx] = a0 * stv + a1 * jd[c];
            }
        }
    }
}

extern "C" void kernel_launch(void* const* d_in, const int* in_sizes, int n_in,
                              void* d_out, int out_size, void* d_ws, size_t ws_size,
                              hipStream_t stream) {
    (void)in_sizes; (void)n_in; (void)out_size; (void)d_ws; (void)ws_size;
    const float* x      = (const float*)d_in[0];
    const float* states = (const float*)d_in[1];
    const float* conv_w = (const float*)d_in[2];
    const float* conv_b = (const float*)d_in[3];
    const float* comp   = (const float*)d_in[4];
    const float* am     = (const float*)d_in[5];
    float* out = (float*)d_out;

    const int nblocks = (NL + TILE - 1) / TILE;  // 4000 blocks of 256 threads
    state_policy_kernel<<<nblocks, 256, 0, stream>>>(x, states, conv_w, conv_b, comp, am, out);
}